// YoloLoss_12910671692458
// MI455X (gfx1250) — compile-verified
//
#include <hip/hip_runtime.h>

// ---------------------------------------------------------------------------
// YOLO loss, MI455X (gfx1250).
// Bandwidth-bound streaming reduction: ~174 MB in / 20 B out -> ~7.5 us floor
// at 23.3 TB/s. Strategy: double-buffered GLOBAL_LOAD_ASYNC_TO_LDS staging
// (ASYNCcnt-tracked, CDNA5 path), compute from LDS, deterministic two-pass
// block reduction (no float atomics).
// ---------------------------------------------------------------------------

#define NCLS             20
#define CELLS            (1024 * 28 * 28)   // 802816
#define CHUNK            128                // cells staged per iteration
#define BLOCKT           128                // threads per block (4 waves)
#define NBLOCKS          784
#define CHUNKS_PER_BLOCK 8                  // 784 * 8 * 128 == CELLS exactly

// Low 32 bits of a generic pointer to __shared__ = workgroup-relative LDS
// byte offset (flat LDS aperture).
__device__ __forceinline__ unsigned lds_off(const void* p) {
    return (unsigned)(unsigned long long)p;
}

// CDNA5 async global->LDS copies (VGLOBAL GV mode, per-lane addresses).
// Tracked with ASYNCcnt; loads complete in order.
__device__ __forceinline__ void async_b64(unsigned lds, const float* g) {
    asm volatile("global_load_async_to_lds_b64 %0, %1, off"
                 :: "v"(lds), "v"(g) : "memory");
}
__device__ __forceinline__ void async_b128(unsigned lds, const float* g) {
    asm volatile("global_load_async_to_lds_b128 %0, %1, off"
                 :: "v"(lds), "v"(g) : "memory");
}

__device__ __forceinline__ float wave_sum(float v) {
    #pragma unroll
    for (int o = 16; o > 0; o >>= 1) v += __shfl_down(v, o, 32);  // wave32
    return v;
}

extern "C" __global__ void __launch_bounds__(BLOCKT)
yolo_main(const float* __restrict__ pred,           // [CELLS][30]
          const float* __restrict__ tbox,           // [CELLS][4]
          const float* __restrict__ tcls,           // [CELLS][20]
          const unsigned char* __restrict__ hasObj, // [CELLS] (jax bool = 1B)
          float4* __restrict__ partials)            // [NBLOCKS]
{
    __shared__ float sPred[2][CHUNK * 30];   // 2 * 15360 B
    __shared__ float sCls [2][CHUNK * NCLS]; // 2 * 10240 B
    __shared__ float sBox [2][CHUNK * 4];    // 2 *  2048 B
    __shared__ float sRed [BLOCKT / 32][4];

    const int tid = threadIdx.x;
    const unsigned chunk0 = blockIdx.x * CHUNKS_PER_BLOCK;

    // Stage one chunk: 21 uniform async instructions per wave.
    auto stage = [&](int buf, unsigned chunk) {
        const unsigned base = chunk * CHUNK;
        const float* gp = pred + (size_t)base * 30;        // 15360 B = 1920 x b64
        const unsigned lp = lds_off(&sPred[buf][0]);
        #pragma unroll
        for (int k = 0; k < 15; ++k) {
            unsigned j = (unsigned)tid + k * BLOCKT;
            async_b64(lp + j * 8u, gp + j * 2);
        }
        const float* gc = tcls + (size_t)base * NCLS;      // 10240 B = 640 x b128
        const unsigned lc = lds_off(&sCls[buf][0]);
        #pragma unroll
        for (int k = 0; k < 5; ++k) {
            unsigned j = (unsigned)tid + k * BLOCKT;
            async_b128(lc + j * 16u, gc + j * 4);
        }
        async_b128(lds_off(&sBox[buf][0]) + (unsigned)tid * 16u,  // 2048 B
                   tbox + (size_t)base * 4 + (size_t)tid * 4);
    };

    stage(0, chunk0);  // prologue

    float aCls = 0.f, aNoObj = 0.f, aReg = 0.f, aCont = 0.f;
    const float invS = 1.f / 28.f;

    #pragma unroll 1
    for (int i = 0; i < CHUNKS_PER_BLOCK; ++i) {
        const int cur = i & 1;
        if (i + 1 < CHUNKS_PER_BLOCK) {
            stage(cur ^ 1, chunk0 + (unsigned)i + 1);      // prefetch next chunk
            // 21 ops just issued for chunk i+1; <=21 outstanding means all of
            // chunk i's 21 async loads (issued earlier, in-order) completed.
            asm volatile("s_wait_asynccnt 21" ::: "memory");
        } else {
            asm volatile("s_wait_asynccnt 0" ::: "memory");
        }
        __syncthreads();  // all waves see buf[cur] fully staged

        const unsigned cell = (chunk0 + (unsigned)i) * CHUNK + (unsigned)tid;
        const float* P  = &sPred[cur][tid * 30];
        const float* C  = &sCls [cur][tid * NCLS];
        const float* TB = &sBox [cur][tid * 4];

        const float m = (hasObj[cell] != 0) ? 1.f : 0.f;

        // ---- class loss: mask * sum_c (pred_cls - tgt_cls)^2
        float cls = 0.f;
        #pragma unroll
        for (int c = 0; c < NCLS; ++c) {
            float d = P[10 + c] - C[c];
            cls = fmaf(d, d, cls);
        }
        aCls += m * cls;

        // ---- no-object loss: 0.5 * (1-m) * (conf0^2 + conf1^2)
        const float conf0 = P[4], conf1 = P[9];
        aNoObj += 0.5f * (1.f - m) * (conf0 * conf0 + conf1 * conf1);

        // ---- target box -> xyxy (x,y scaled by 1/S; w,h raw)
        const float tx = TB[0], ty = TB[1], tw = TB[2], th = TB[3];
        const float tcx = tx * invS, tcy = ty * invS;
        const float tx1 = tcx - 0.5f * tw, ty1 = tcy - 0.5f * th;
        const float tx2 = tcx + 0.5f * tw, ty2 = tcy + 0.5f * th;
        const float areaT = (tx2 - tx1) * (ty2 - ty1);

        // ---- IoU of both predicted boxes vs target
        float iou[2];
        #pragma unroll
        for (int b = 0; b < 2; ++b) {
            const float bx = P[b * 5 + 0], by = P[b * 5 + 1];
            const float bw = P[b * 5 + 2], bh = P[b * 5 + 3];
            const float cx = bx * invS, cy = by * invS;
            const float x1 = cx - 0.5f * bw, y1 = cy - 0.5f * bh;
            const float x2 = cx + 0.5f * bw, y2 = cy + 0.5f * bh;
            const float wI = fmaxf(fminf(x2, tx2) - fmaxf(x1, tx1), 0.f);
            const float hI = fmaxf(fminf(y2, ty2) - fmaxf(y1, ty1), 0.f);
            const float inter = wI * hI;
            const float areaP = (x2 - x1) * (y2 - y1);
            iou[b] = inter / (areaP + areaT - inter);
        }
        // jnp.argmax returns first max -> box 1 wins only on strict greater
        const int   bi      = (iou[1] > iou[0]) ? 1 : 0;
        const float bestIou = (iou[1] > iou[0]) ? iou[1] : iou[0];
        const float bx = P[bi * 5 + 0], by = P[bi * 5 + 1];
        const float bw = P[bi * 5 + 2], bh = P[bi * 5 + 3];
        const float bc = P[bi * 5 + 4];

        // ---- regression loss (raw coords for center error, sqrt for size)
        const float dx = bx - tx, dy = by - ty;
        const float centerErr = dx * dx + dy * dy;
        const bool obj = (m != 0.f);
        const float pw = sqrtf(obj ? fmaxf(bw, 0.f) : 1.f);
        const float ph = sqrtf(obj ? fmaxf(bh, 0.f) : 1.f);
        const float sw = sqrtf(obj ? tw : 1.f);
        const float sh = sqrtf(obj ? th : 1.f);
        const float sizeErr = (pw - sw) * (pw - sw) + (ph - sh) * (ph - sh);
        aReg += 5.f * m * (centerErr + sizeErr);

        // ---- containment loss: mask * (best_conf - best_iou)^2
        const float dc = bc - bestIou;
        aCont += m * dc * dc;

        __syncthreads();  // done reading buf[cur]; safe to overwrite at i+2
    }

    // ---- deterministic block reduction -> one float4 partial per block
    aCls   = wave_sum(aCls);
    aNoObj = wave_sum(aNoObj);
    aReg   = wave_sum(aReg);
    aCont  = wave_sum(aCont);
    const int lane = tid & 31, wid = tid >> 5;
    if (lane == 0) {
        sRed[wid][0] = aCls;  sRed[wid][1] = aNoObj;
        sRed[wid][2] = aReg;  sRed[wid][3] = aCont;
    }
    __syncthreads();
    if (tid == 0) {
        float c = 0.f, n = 0.f, r = 0.f, t = 0.f;
        #pragma unroll
        for (int w = 0; w < BLOCKT / 32; ++w) {
            c += sRed[w][0]; n += sRed[w][1]; r += sRed[w][2]; t += sRed[w][3];
        }
        partials[blockIdx.x] = make_float4(c, n, r, t);  // (cls, noobj, reg, contain)
    }
}

extern "C" __global__ void __launch_bounds__(256)
yolo_finalize(const float4* __restrict__ partials, float* __restrict__ out)
{
    __shared__ float sRed[8][4];
    const int tid = threadIdx.x;
    float c = 0.f, n = 0.f, r = 0.f, t = 0.f;
    for (int i = tid; i < NBLOCKS; i += 256) {
        float4 p = partials[i];
        c += p.x; n += p.y; r += p.z; t += p.w;
    }
    c = wave_sum(c); n = wave_sum(n); r = wave_sum(r); t = wave_sum(t);
    const int lane = tid & 31, wid = tid >> 5;
    if (lane == 0) { sRed[wid][0] = c; sRed[wid][1] = n; sRed[wid][2] = r; sRed[wid][3] = t; }
    __syncthreads();
    if (tid == 0) {
        float C = 0.f, N = 0.f, R = 0.f, T = 0.f;
        #pragma unroll
        for (int w = 0; w < 8; ++w) {
            C += sRed[w][0]; N += sRed[w][1]; R += sRed[w][2]; T += sRed[w][3];
        }
        const float total = (C + N + T + R) * (1.f / 1024.f);
        out[0] = total;  // total_loss
        out[1] = R;      // reg_loss
        out[2] = T;      // contain_loss
        out[3] = N;      // no_obj_loss
        out[4] = C;      // cls_loss
    }
}

extern "C" void kernel_launch(void* const* d_in, const int* in_sizes, int n_in,
                              void* d_out, int out_size, void* d_ws, size_t ws_size,
                              hipStream_t stream)
{
    (void)in_sizes; (void)n_in; (void)out_size; (void)ws_size;
    const float*         pred   = (const float*)d_in[0];         // [1024,28,28,30]
    const float*         tbox   = (const float*)d_in[1];         // [1024,28,28,4]
    const float*         tcls   = (const float*)d_in[2];         // [1024,28,28,20]
    const unsigned char* hasObj = (const unsigned char*)d_in[3]; // [1024,28,28] bool
    float4* partials = (float4*)d_ws;                            // 784 * 16 B

    yolo_main<<<NBLOCKS, BLOCKT, 0, stream>>>(pred, tbox, tcls, hasObj, partials);
    yolo_finalize<<<1, 256, 0, stream>>>(partials, (float*)d_out);
}